// OldTensorProductConvLayer_18760417149590
// MI455X (gfx1250) — compile-verified
//
#include <hip/hip_runtime.h>
#include <math.h>

// ---------------------------------------------------------------------------
// Fused equivariant tensor-product conv layer for MI455X (gfx1250, wave32).
// w = mish(edge_attr@W1+b1)@W2+b2 is generated chunk-by-chunk with
// V_WMMA_F32_16X16X4_F32 and consumed in-register (never hits HBM).
// fc2 B-chunks are staged into LDS by the Tensor Data Mover (double-buffered,
// overlapped with WMMA compute, bank-conflict-free via TDM padding).
// ---------------------------------------------------------------------------

typedef float v2f __attribute__((ext_vector_type(2)));
typedef float v8f __attribute__((ext_vector_type(8)));
typedef unsigned int u32x4 __attribute__((ext_vector_type(4)));
typedef int i32x4 __attribute__((ext_vector_type(4)));
typedef int i32x8 __attribute__((ext_vector_type(8)));

#define N_NODES 10000
#define N_EDGES 100000
#define N_EF    64
#define NS      32
#define NV      8
#define D_NODE  56            // NS + 3*NV
#define WNUM    1600          // 1024 + 64 + 256 + 256
#define NCHUNK  (WNUM / 16)   // 100 fc2 column chunks
#define A_SCALE 0.15811388300841897f   // 1/sqrt(NS+NV) = 1/sqrt(40)
#define RSQRT3  0.57735026918962576f
#define WAVES   2             // waves per block
#define EPW     32            // edges per wave (2 M-tiles)
#define BROW    66            // LDS row stride for B chunk (TDM pad: 64+2)

__device__ __forceinline__ float mish_f(float x) {
  // mish(x) = x * tanh(softplus(x));  tanh(log1p(t)) = (t^2+2t)/(t^2+2t+2)
  float t = __expf(x);
  float q = t * t + 2.f * t;
  float r = x * q / (q + 2.f);
  return (x > 20.f) ? x : r;
}

__device__ __forceinline__ v8f wmma_f32(v2f a, v2f b, v8f c) {
  // D = A(16x4 f32) * B(4x16 f32) + C(16x16 f32)
  return __builtin_amdgcn_wmma_f32_16x16x4_f32(false, a, false, b, (short)0, c,
                                               false, false);
}

// ---- TDM: DMA one contiguous 4KB fc2 B-chunk (16 cols x 64 K) into LDS ----
// group1: data_size=4B(2), pad_enable, pad_interval=64DW(5), pad_amount=2DW(1)
//         tensor_dim0=1024, tensor_dim1=1, tile_dim0=1024, tile_dim1=1,
//         dim0_stride=1024  => LDS rows land at stride 66 (bank-conflict-free)
__device__ __forceinline__ void tdm_load_chunk(const float* gsrc,
                                               float* lds_dst) {
  unsigned long long ga = (unsigned long long)(uintptr_t)gsrc;
  unsigned int la = (unsigned int)(uintptr_t)lds_dst;  // addr[31:0] = LDS off
  u32x4 g0;
  g0[0] = 1u;                                           // count=1 (user D#)
  g0[1] = la;                                           // lds_addr
  g0[2] = (unsigned int)ga;                             // global_addr[31:0]
  g0[3] = ((unsigned int)(ga >> 32) & 0x01FFFFFFu) | 0x80000000u; // hi|type=2
  i32x8 g1 = {0x03520000, 0x04000000, 0x00010000, 0x04000000,
              1, 0x400, 0, 0};
  i32x4 gz4 = {0, 0, 0, 0};
#if __has_include(<hip/amd_detail/amd_gfx1250_TDM.h>)
  i32x8 gz8 = {0, 0, 0, 0, 0, 0, 0, 0};
  __builtin_amdgcn_tensor_load_to_lds(g0, g1, gz4, gz4, gz8, 0);
#else
  __builtin_amdgcn_tensor_load_to_lds(g0, g1, gz4, gz4, 0);
#endif
}

// ---- one-time weight transpose: wT[col][k] = w[k][col] --------------------
__global__ __launch_bounds__(256)
void transpose_w(const float* __restrict__ fc1_w, const float* __restrict__ fc2_w,
                 float* __restrict__ wT1, float* __restrict__ wT2) {
  int idx0 = blockIdx.x * blockDim.x + threadIdx.x;
  int stride = gridDim.x * blockDim.x;
  for (int idx = idx0; idx < N_EF * N_EF; idx += stride) {
    int k = idx >> 6, c = idx & 63;
    wT1[c * N_EF + k] = fc1_w[idx];
  }
  for (int idx = idx0; idx < N_EF * WNUM; idx += stride) {
    int k = idx / WNUM, c = idx % WNUM;
    wT2[c * N_EF + k] = fc2_w[idx];
  }
}

__global__ __launch_bounds__(WAVES * 32)
void tpconv_edges(const float* __restrict__ node_attr,
                  const int*   __restrict__ edge_index,
                  const float* __restrict__ edge_attr,
                  const float* __restrict__ edge_sh,
                  const float* __restrict__ wT1,
                  const float* __restrict__ fc1_b,
                  const float* __restrict__ wT2,
                  const float* __restrict__ fc2_b,
                  float* __restrict__ accum,   // [N_NODES][56] segment sums
                  float* __restrict__ cnt)     // [N_NODES]
{
  __shared__ __align__(16) float btile[2][16 * BROW];  // TDM-staged B chunks
  __shared__ float hbuf[WAVES][EPW][N_EF + 2]; // mish(fc1), padded (8B align)
  __shared__ float sL   [WAVES][EPW][NS];      // raw scalars of x[dst]
  __shared__ float sHatL[WAVES][EPW][NS];      // s * sh0
  __shared__ float vL   [WAVES][EPW][3 * NV];  // raw vectors, u-major ch-minor
  __shared__ float sh0L [WAVES][EPW];
  __shared__ float sh1L [WAVES][EPW][3];
  __shared__ float dotL [WAVES][EPW][NV];      // <v_u, sh1>/sqrt(3)
  __shared__ float accvL[WAVES][EPW][3 * NV];  // out_v accumulators
  __shared__ int   srcL [WAVES][EPW];

  const int tid  = threadIdx.x;
  const int wave = tid >> 5;
  const int lane = tid & 31;
  const int n16  = lane & 15;      // WMMA N / M index within half
  const int hf   = lane >> 4;      // lane half
  const int e0   = (blockIdx.x * WAVES + wave) * EPW;

  const v2f zero2 = {0.f, 0.f};
  const v8f zero8 = {0.f, 0.f, 0.f, 0.f, 0.f, 0.f, 0.f, 0.f};

  // ---------------- per-edge staging ----------------
  {
    int e = e0 + lane;                       // 32 edges, 32 lanes
    bool valid = e < N_EDGES;
    srcL[wave][lane]    = valid ? edge_index[e] : 0;
    sh0L[wave][lane]    = valid ? edge_sh[e * 4 + 0] : 0.f;
    sh1L[wave][lane][0] = valid ? edge_sh[e * 4 + 1] : 0.f;
    sh1L[wave][lane][1] = valid ? edge_sh[e * 4 + 2] : 0.f;
    sh1L[wave][lane][2] = valid ? edge_sh[e * 4 + 3] : 0.f;
  }
  for (int idx = lane; idx < EPW * D_NODE; idx += 32) {
    int i = idx / D_NODE, c = idx % D_NODE;
    int e = e0 + i;
    float val = 0.f;
    if (e < N_EDGES) {
      int dst = edge_index[N_EDGES + e];
      val = node_attr[(size_t)dst * D_NODE + c];
    }
    if (c < NS) sL[wave][i][c] = val;
    else        vL[wave][i][c - NS] = val;
  }
  __syncthreads();
  for (int idx = lane; idx < EPW * NS; idx += 32) {
    int i = idx >> 5, u = idx & 31;
    sHatL[wave][i][u] = sL[wave][i][u] * sh0L[wave][i];
  }
  for (int idx = lane; idx < EPW * NV; idx += 32) {
    int i = idx >> 3, u = idx & 7;
    dotL[wave][i][u] = (vL[wave][i][u * 3 + 0] * sh1L[wave][i][0] +
                        vL[wave][i][u * 3 + 1] * sh1L[wave][i][1] +
                        vL[wave][i][u * 3 + 2] * sh1L[wave][i][2]) * RSQRT3;
  }
  for (int idx = lane; idx < EPW * 3 * NV; idx += 32)
    accvL[wave][idx / 24][idx % 24] = 0.f;
  __syncthreads();

  // ---------------- fc1: h = mish(edge_attr @ fc1_w + b1), per M-tile ------
#pragma unroll
  for (int tt = 0; tt < 2; ++tt) {
    v2f a1[16];
    int e = e0 + tt * 16 + n16;
    bool valid = e < N_EDGES;
    const v2f* ea2 =
        (const v2f*)(edge_attr + (size_t)(valid ? e : 0) * N_EF);
#pragma unroll
    for (int k = 0; k < 16; ++k) a1[k] = valid ? ea2[2 * k + hf] : zero2;

#pragma unroll
    for (int j = 0; j < 4; ++j) {
      v8f c = zero8;
      const v2f* bp = (const v2f*)(wT1 + (size_t)(j * 16 + n16) * N_EF);
#pragma unroll
      for (int k = 0; k < 16; ++k) c = wmma_f32(a1[k], bp[2 * k + hf], c);
      float bias = fc1_b[j * 16 + n16];
#pragma unroll
      for (int r = 0; r < 8; ++r) {
        int m = tt * 16 + r + 8 * hf;  // C layout: lanes16-31 hold M=r+8
        hbuf[wave][m][j * 16 + n16] = mish_f(c[r] + bias);
      }
    }
  }
  __syncthreads();

  // fc2 A-operands (2 x 16x64 h tiles) resident in registers for all chunks
  v2f ha[2][16];
#pragma unroll
  for (int t = 0; t < 2; ++t)
#pragma unroll
    for (int k = 0; k < 16; ++k)
      ha[t][k] = *(const v2f*)&hbuf[wave][t * 16 + n16][4 * k + 2 * hf];

  v8f accLo[2] = {zero8, zero8};  // out_s cols 0..15 per tile
  v8f accHi[2] = {zero8, zero8};  // out_s cols 16..31 per tile

  // ---------------- fc2 chunks via TDM double-buffer ----------------------
  if (wave == 0) {
    tdm_load_chunk(wT2, &btile[0][0]);            // prologue: chunk 0
    __builtin_amdgcn_s_wait_tensorcnt(0);
  }
  __syncthreads();

  for (int cblk = 0; cblk < NCHUNK; ++cblk) {
    const int cur = cblk & 1;
    if (wave == 0 && cblk + 1 < NCHUNK)           // overlap DMA with compute
      tdm_load_chunk(wT2 + (size_t)(cblk + 1) * 1024, &btile[cur ^ 1][0]);

    v2f bb[16];
    const v2f* bp = (const v2f*)&btile[cur][n16 * BROW];
#pragma unroll
    for (int k = 0; k < 16; ++k) bb[k] = bp[2 * k + hf];
    float bias = fc2_b[cblk * 16 + n16];

#pragma unroll
    for (int t = 0; t < 2; ++t) {
      v8f c = zero8;
#pragma unroll
      for (int k = 0; k < 16; ++k) c = wmma_f32(ha[t][k], bb[k], c);

      if (cblk < 64) {                          // ---- w1: out_s += (s*sh0)@w1
        int u = cblk >> 1;
        if (cblk & 1) {
#pragma unroll
          for (int r = 0; r < 8; ++r) {
            int m = t * 16 + r + 8 * hf;
            accHi[t][r] += (A_SCALE * sHatL[wave][m][u]) * (c[r] + bias);
          }
        } else {
#pragma unroll
          for (int r = 0; r < 8; ++r) {
            int m = t * 16 + r + 8 * hf;
            accLo[t][r] += (A_SCALE * sHatL[wave][m][u]) * (c[r] + bias);
          }
        }
      } else if (cblk < 68) {                   // ---- w2: out_v += (v*sh0)@w2
        int u  = 2 * (cblk - 64) + (n16 >> 3);
        int wc = n16 & 7;
#pragma unroll
        for (int r = 0; r < 8; ++r) {
          int m = t * 16 + r + 8 * hf;
          float val = A_SCALE * sh0L[wave][m] * (c[r] + bias);
          atomicAdd(&accvL[wave][m][wc * 3 + 0], vL[wave][m][u * 3 + 0] * val);
          atomicAdd(&accvL[wave][m][wc * 3 + 1], vL[wave][m][u * 3 + 1] * val);
          atomicAdd(&accvL[wave][m][wc * 3 + 2], vL[wave][m][u * 3 + 2] * val);
        }
      } else if (cblk < 84) {                   // ---- w3: out_v += (s x sh1)@w3
        int u  = 2 * (cblk - 68) + (n16 >> 3);
        int wc = n16 & 7;
#pragma unroll
        for (int r = 0; r < 8; ++r) {
          int m = t * 16 + r + 8 * hf;
          float val = A_SCALE * sL[wave][m][u] * (c[r] + bias);
          atomicAdd(&accvL[wave][m][wc * 3 + 0], sh1L[wave][m][0] * val);
          atomicAdd(&accvL[wave][m][wc * 3 + 1], sh1L[wave][m][1] * val);
          atomicAdd(&accvL[wave][m][wc * 3 + 2], sh1L[wave][m][2] * val);
        }
      } else {                                  // ---- w4: out_s += dot@w4
        int u = (cblk - 84) >> 1;
        if (cblk & 1) {
#pragma unroll
          for (int r = 0; r < 8; ++r) {
            int m = t * 16 + r + 8 * hf;
            accHi[t][r] += (A_SCALE * dotL[wave][m][u]) * (c[r] + bias);
          }
        } else {
#pragma unroll
          for (int r = 0; r < 8; ++r) {
            int m = t * 16 + r + 8 * hf;
            accLo[t][r] += (A_SCALE * dotL[wave][m][u]) * (c[r] + bias);
          }
        }
      }
    }

    if (wave == 0 && cblk + 1 < NCHUNK)
      __builtin_amdgcn_s_wait_tensorcnt(0);     // publish next buffer
    __syncthreads();
  }

  // ---------------- scatter (segment_sum over edge_src) ----------------
#pragma unroll
  for (int t = 0; t < 2; ++t)
#pragma unroll
    for (int r = 0; r < 8; ++r) {
      int m = t * 16 + r + 8 * hf;
      int e = e0 + m;
      if (e < N_EDGES) {
        int src = srcL[wave][m];
        atomicAdd(&accum[(size_t)src * D_NODE + n16],      accLo[t][r]);
        atomicAdd(&accum[(size_t)src * D_NODE + 16 + n16], accHi[t][r]);
      }
    }
  for (int idx = lane; idx < EPW * 3 * NV; idx += 32) {
    int i = idx / 24, j = idx % 24;
    int e = e0 + i;
    if (e < N_EDGES)
      atomicAdd(&accum[(size_t)srcL[wave][i] * D_NODE + NS + j],
                accvL[wave][i][j]);
  }
  {
    int e = e0 + lane;
    if (e < N_EDGES) atomicAdd(&cnt[srcL[wave][lane]], 1.0f);
  }
}

// ---- mean over edges, residual, BN statistics (block-reduced) -------------
__global__ __launch_bounds__(256)
void node_stats(const float* __restrict__ accum, const float* __restrict__ cnt,
                const float* __restrict__ node_attr, float* __restrict__ outp,
                float* __restrict__ stats) {
  __shared__ float bins[72];   // [0..31]=sum_s  [32..63]=sumsq_s  [64..71]=sumsq_v
  for (int i = threadIdx.x; i < 72; i += blockDim.x) bins[i] = 0.f;
  __syncthreads();
  for (int idx = blockIdx.x * blockDim.x + threadIdx.x; idx < N_NODES * D_NODE;
       idx += gridDim.x * blockDim.x) {
    int n = idx / D_NODE, c = idx % D_NODE;
    float v = accum[idx] / fmaxf(cnt[n], 1.f) + node_attr[idx];
    outp[idx] = v;
    if (c < NS) {
      atomicAdd(&bins[c], v);
      atomicAdd(&bins[32 + c], v * v);
    } else {
      atomicAdd(&bins[64 + (c - NS) / 3], v * v);
    }
  }
  __syncthreads();
  for (int i = threadIdx.x; i < 72; i += blockDim.x) atomicAdd(&stats[i], bins[i]);
}

// ---- apply batch norm ------------------------------------------------------
__global__ __launch_bounds__(256)
void bn_apply(const float* __restrict__ outp, const float* __restrict__ stats,
              const float* __restrict__ g_s, const float* __restrict__ b_s,
              const float* __restrict__ g_v, float* __restrict__ out) {
  int idx = blockIdx.x * blockDim.x + threadIdx.x;
  if (idx >= N_NODES * D_NODE) return;
  int c = idx % D_NODE;
  float x = outp[idx];
  if (c < NS) {
    float mu  = stats[c] * (1.f / N_NODES);
    float var = stats[32 + c] * (1.f / N_NODES) - mu * mu;
    out[idx] = (x - mu) * rsqrtf(var + 1e-4f) * g_s[c] + b_s[c];
  } else {
    int u = (c - NS) / 3;
    float vn = stats[64 + u] * (1.f / (3.f * N_NODES));
    out[idx] = x * rsqrtf(vn + 1e-4f) * g_v[u];
  }
}

extern "C" void kernel_launch(void* const* d_in, const int* in_sizes, int n_in,
                              void* d_out, int out_size, void* d_ws, size_t ws_size,
                              hipStream_t stream) {
  const float* node_attr = (const float*)d_in[0];
  const int*   edge_index = (const int*)d_in[1];
  const float* edge_attr = (const float*)d_in[2];
  const float* edge_sh   = (const float*)d_in[3];
  const float* fc1_w = (const float*)d_in[4];
  const float* fc1_b = (const float*)d_in[5];
  const float* fc2_w = (const float*)d_in[6];
  const float* fc2_b = (const float*)d_in[7];
  const float* g_s = (const float*)d_in[8];
  const float* b_s = (const float*)d_in[9];
  const float* g_v = (const float*)d_in[10];

  float* stats = (float*)d_ws;                      // 72
  float* cnt   = stats + 72;                        // N_NODES
  float* accum = cnt + N_NODES;                     // N_NODES*56
  float* outp  = accum + (size_t)N_NODES * D_NODE;  // N_NODES*56
  float* wT1   = outp + (size_t)N_NODES * D_NODE;   // 64*64
  float* wT2   = wT1 + N_EF * N_EF;                 // 64*1600

  hipMemsetAsync(d_ws, 0,
                 (size_t)(72 + N_NODES + (size_t)N_NODES * D_NODE) * sizeof(float),
                 stream);

  transpose_w<<<128, 256, 0, stream>>>(fc1_w, fc2_w, wT1, wT2);

  int eblocks = (N_EDGES + EPW * WAVES - 1) / (EPW * WAVES);
  tpconv_edges<<<eblocks, 32 * WAVES, 0, stream>>>(
      node_attr, edge_index, edge_attr, edge_sh, wT1, fc1_b, wT2, fc2_b,
      accum, cnt);

  node_stats<<<256, 256, 0, stream>>>(accum, cnt, node_attr, outp, stats);

  bn_apply<<<(N_NODES * D_NODE + 255) / 256, 256, 0, stream>>>(
      outp, stats, g_s, b_s, g_v, (float*)d_out);
}